// Seesaw_Conv_35089882808337
// MI455X (gfx1250) — compile-verified
//
#include <hip/hip_runtime.h>

// Problem constants (from reference): B=8, C=3, H=W=512, KS=3, K=9
#define BI 8
#define CI 3
#define HI 512
#define WI 512
#define HW (HI*WI)

#define TILE_PX 256      // pixels (one row segment) per block; W/TILE_PX == 2
#define SBW     260      // im2col row stride (260 % 64 == 4 -> no bank conflicts)

typedef float v2f __attribute__((ext_vector_type(2)));
typedef float v8f __attribute__((ext_vector_type(8)));

__device__ __forceinline__ float sigmoidf_(float z) {
    return 1.0f / (1.0f + __expf(-z));
}

// Fused: conv3x3(+bias,sigmoid) via V_WMMA_F32_16X16X4_F32 -> 9-point bilinear
// gather -> grouped 1x1 conv (+bias, ReLU).
//
// K ordering for the GEMM is permuted to K' = tx*9 + (c*3 + ty) so that the
// im2col matrix sB[K'][pixel] gives B-operand LDS addresses that are affine in
// the k-step: per-iteration fetches use only immediate ds offsets.
__global__ __launch_bounds__(256)
void seesaw_fused_kernel(const float* __restrict__ x,
                         const float* __restrict__ w1,
                         const float* __restrict__ b1,
                         const float* __restrict__ w2,
                         const float* __restrict__ b2,
                         float* __restrict__ out) {
    // A matrix (16 out-ch x 28 K', zero padded: rows 8..15 and col 27 are 0)
    __shared__ float sA[16*28];
    // Explicit im2col: 27 data rows (K') x 256 pixels (stride 260) + 1 zero row
    __shared__ float sB[28*SBW];
    __shared__ float sW2[27];
    __shared__ float sB1[8];
    __shared__ float sB2[3];

    const int tid  = threadIdx.x;
    const int lane = tid & 31;
    const int wave = tid >> 5;

    const int blk  = blockIdx.x;          // grid = B * H * (W/TILE_PX) = 8192
    const int xblk = blk & 1;
    const int y    = (blk >> 1) & (HI - 1);
    const int b    = blk >> 10;           // blk / (2*512)
    const int x0   = xblk * TILE_PX;

    // ---- stage conv weights as WMMA-A (with K' permutation) ----
    for (int i = tid; i < 16*28; i += 256) {
        int mm = i / 28, kp = i - mm*28;
        float v = 0.0f;
        if (mm < 8 && kp < 27) {
            int tx = kp / 9, rem = kp - 9*tx;
            int c  = rem / 3, ty = rem - 3*c;
            // w1 flat [8][3][3][3] -> mm*27 + c*9 + ty*3 + tx
            v = w1[mm*27 + c*9 + ty*3 + tx];
        }
        sA[i] = v;
    }
    if (tid < 27)                 sW2[tid]      = w2[tid];
    if (tid >= 32 && tid < 40)    sB1[tid - 32] = b1[tid - 32];
    if (tid >= 64 && tid < 67)    sB2[tid - 64] = b2[tid - 64];

    // ---- build im2col sB directly from global (unrolled, division-free) ----
    const float* xb = x + (size_t)b * CI * HW;
#pragma unroll
    for (int rr = 0; rr < 27; ++rr) {
        const int tx  = rr / 9;           // compile-time
        const int rem = rr % 9;
        const int c   = rem / 3;
        const int ty  = rem % 3;
        const int gyr = y - 1 + ty;               // uniform
        const int gxc = x0 - 1 + tid + tx;        // per-thread
        const bool ok = ((unsigned)gyr < HI) & ((unsigned)gxc < WI);
        const int  yc = min(max(gyr, 0), HI - 1);
        const int  xc = min(max(gxc, 0), WI - 1);
        const float v = xb[c*HW + yc*WI + xc];    // L0/L2 hit (shared lines)
        sB[rr*SBW + tid] = ok ? v : 0.0f;
    }
    sB[27*SBW + tid] = 0.0f;                      // K'==27 pad row
    __syncthreads();

    // ---- conv3x3 as fp32 WMMA: D[16ch x 16px] = sum_s A[16x4] * B[4x16] ----
    // A layout (ISA 7.12.2, 32-bit A 16x4): lanes {m, m+16} hold row m;
    //   VGPR0 = K{4s+0 | 4s+2}, VGPR1 = K{4s+1 | 4s+3}  (h = lane>>4)
    // B layout (row striped across lanes): lane n holds column n;
    //   VGPR0 = K{4s+0 | 4s+1}, VGPR1 = K{4s+2 | 4s+3}
    const int m  = lane & 15;
    const int h  = lane >> 4;
    const float* aA  = sA + m*28 + 2*h;           // imm offsets 16s bytes
    const float* bB0 = sB + h*SBW + wave*32 + m;  // imm offsets 4s*SBW*4 bytes
    const float* bB1 = bB0 + 16;

    v8f acc0 = {};
    v8f acc1 = {};
#pragma unroll
    for (int s = 0; s < 7; ++s) {
        v2f a;
        a.x = aA[4*s];
        a.y = aA[4*s + 1];

        v2f bt0, bt1;
        bt0.x = bB0[(4*s    )*SBW];
        bt0.y = bB0[(4*s + 2)*SBW];
        bt1.x = bB1[(4*s    )*SBW];
        bt1.y = bB1[(4*s + 2)*SBW];

        // (neg_a, A, neg_b, B, c_mod, C, reuse_a, reuse_b)
        acc0 = __builtin_amdgcn_wmma_f32_16x16x4_f32(false, a, false, bt0,
                                                     (short)0, acc0, false, false);
        acc1 = __builtin_amdgcn_wmma_f32_16x16x4_f32(false, a, false, bt1,
                                                     (short)0, acc1, false, false);
    }

    // ---- redistribute D so every lane owns one pixel's 8 channels ----
    // D layout: VGPR r, lanes 0-15 -> (ch r, px n); lanes 16-31 -> ch 8+r (pad).
    float conv8[8];
    const int src = lane & 15;
#pragma unroll
    for (int r = 0; r < 8; ++r) {
        float v0 = __shfl(acc0[r], src, 32);
        float v1 = __shfl(acc1[r], src, 32);
        conv8[r] = (lane < 16) ? v0 : v1;
    }

    const int xp = x0 + wave*32 + lane;   // this lane's pixel column

    float sj[8];
#pragma unroll
    for (int j = 0; j < 8; ++j) sj[j] = sigmoidf_(conv8[j] + sB1[j]);

    // base grid: linspace(0,1): base0 = y/(H-1), base1 = x/(W-1)
    const float by = (float)y  * (1.0f/511.0f);
    const float bx = (float)xp * (1.0f/511.0f);

    // 9 points: 4 offsets, center, 4 mirrored (2 - off)
    float p0[9], p1[9];
#pragma unroll
    for (int i = 0; i < 4; ++i) { p0[i] = by + sj[2*i]; p1[i] = bx + sj[2*i+1]; }
    p0[4] = by; p1[4] = bx;
#pragma unroll
    for (int i = 0; i < 4; ++i) {
        p0[5+i] = by + (2.0f - sj[2*i]);
        p1[5+i] = bx + (2.0f - sj[2*i+1]);
    }

    // ---- bilinear gather (zeros padding, align_corners=False) + grouped 1x1 ----
    // Predicated (no EXEC churn): clamp index, zero the weight when OOB.
    // Reference quirk preserved: ix comes from coords[...,0] (built from base-y).
    float acc[3] = {0.0f, 0.0f, 0.0f};
#pragma unroll
    for (int kpt = 0; kpt < 9; ++kpt) {
        const float ix = (p0[kpt] + 1.0f) * 256.0f - 0.5f;  // ((g+1)*W - 1)/2
        const float iy = (p1[kpt] + 1.0f) * 256.0f - 0.5f;
        const float fx = floorf(ix), fy = floorf(iy);
        const float wx1 = ix - fx, wy1 = iy - fy;
        const float wx0 = 1.0f - wx1, wy0 = 1.0f - wy1;
        const int xi0 = (int)fx, yi0 = (int)fy;

        const int   cyi[2] = { yi0, yi0 + 1 };
        const int   cxi[2] = { xi0, xi0 + 1 };
        const float wyf[2] = { wy0, wy1 };
        const float wxf[2] = { wx0, wx1 };

        float s0 = 0.0f, s1 = 0.0f, s2 = 0.0f;
#pragma unroll
        for (int dy = 0; dy < 2; ++dy) {
#pragma unroll
            for (int dx = 0; dx < 2; ++dx) {
                const int yy = cyi[dy], xx = cxi[dx];
                const bool ok = ((unsigned)yy < HI) & ((unsigned)xx < WI);
                const int  yc = min(max(yy, 0), HI - 1);
                const int  xc = min(max(xx, 0), WI - 1);
                const float w = ok ? (wyf[dy] * wxf[dx]) : 0.0f;
                const int idx = yc*WI + xc;
                s0 += w * xb[idx];              // L2-resident gathers
                s1 += w * xb[HW + idx];
                s2 += w * xb[2*HW + idx];
            }
        }
        // grouped conv: flat channel j = kpt*3 + c indexes w2 flat [3][9]
        acc[kpt/3] += sW2[3*kpt + 0]*s0 + sW2[3*kpt + 1]*s1 + sW2[3*kpt + 2]*s2;
    }

    float* ob = out + (size_t)b * CI * HW + (size_t)y * WI + xp;
#pragma unroll
    for (int g = 0; g < 3; ++g) {
        float o = acc[g] + sB2[g];
        ob[(size_t)g * HW] = fmaxf(o, 0.0f);   // coalesced NCHW store
    }
}

extern "C" void kernel_launch(void* const* d_in, const int* in_sizes, int n_in,
                              void* d_out, int out_size, void* d_ws, size_t ws_size,
                              hipStream_t stream) {
    const float* x  = (const float*)d_in[0];  // [8,3,512,512]
    const float* w1 = (const float*)d_in[1];  // [8,3,3,3]
    const float* b1 = (const float*)d_in[2];  // [8]
    const float* w2 = (const float*)d_in[3];  // [3,9]
    const float* b2 = (const float*)d_in[4];  // [3]
    float* out = (float*)d_out;               // [8,3,512,512]

    dim3 grid(BI * HI * (WI / TILE_PX));      // 8192 blocks
    seesaw_fused_kernel<<<grid, 256, 0, stream>>>(x, w1, b1, w2, b2, out);
}